// SEP_G_4492535791675
// MI455X (gfx1250) — compile-verified
//
#include <hip/hip_runtime.h>
#include <stdint.h>

// ---------------------------------------------------------------------------
// SEP-G GNN pipeline for gfx1250 (MI455X).
//  - fp32 WMMA (V_WMMA_F32_16X16X4_F32) for all dense GEMMs
//  - Tensor Data Mover (tensor_load_to_lds) double-buffers A/B tiles into LDS
//    (s_wait_tensorcnt(2) overlaps next-chunk DMA with current-chunk WMMAs)
//  - sparse aggregation / pooling via wave-per-row float4 atomics
// ---------------------------------------------------------------------------

typedef __attribute__((ext_vector_type(2))) float v2f;
typedef __attribute__((ext_vector_type(8))) float v8f;
typedef __attribute__((ext_vector_type(4))) unsigned int u32x4;
typedef __attribute__((ext_vector_type(8))) int i32x8;
typedef __attribute__((ext_vector_type(4))) int i32x4;

#define LEAKY_SLOPE 0.01f

// ---------------------------------------------------------------------------
// TDM: issue a 2D tile load (global -> LDS), fp32 elements.
//   tile = rows x row_len; tensor extents (from tile origin) td0 x td1 give
//   hardware zero-fill for OOB; row stride = stride_elems.
// D# packing per CDNA5 ISA 8.3/8.4 (count=1, type=2 image, data_size=2 -> 4B).
// ---------------------------------------------------------------------------
__device__ __forceinline__ void tdm_load_2d_f32(
    uint32_t lds_byte_addr, const float* gaddr,
    uint32_t row_len, uint32_t rows,
    uint32_t td0, uint32_t td1, uint64_t stride_elems)
{
  uint64_t ga = (uint64_t)(uintptr_t)gaddr;
  u32x4 g0;
  g0.x = 1u;                                                  // count=1 (valid user D#)
  g0.y = lds_byte_addr;                                       // lds_addr
  g0.z = (uint32_t)ga;                                        // global_addr[31:0]
  g0.w = (uint32_t)((ga >> 32) & 0x1FFFFFFu) | (2u << 30);    // global_addr[56:32] | type=2
  i32x8 g1;
  g1[0] = (int)(2u << 16);                                    // data_size=2 (4 bytes)
  g1[1] = (int)((td0 & 0xFFFFu) << 16);                       // tensor_dim0[15:0]
  g1[2] = (int)((td0 >> 16) | ((td1 & 0xFFFFu) << 16));       // tensor_dim0[31:16]|tensor_dim1[15:0]
  g1[3] = (int)((td1 >> 16) | (row_len << 16));               // tensor_dim1[31:16]|tile_dim0
  g1[4] = (int)(rows & 0xFFFFu);                              // tile_dim1 (tile_dim2=0)
  g1[5] = (int)(uint32_t)(stride_elems & 0xFFFFFFFFu);        // tensor_dim0_stride[31:0]
  g1[6] = (int)(uint32_t)((stride_elems >> 32) & 0xFFFFu);    // stride[47:32]
  g1[7] = 0;
  i32x4 z4 = {0, 0, 0, 0};
#if __has_include(<hip/amd_detail/amd_gfx1250_TDM.h>)
  i32x8 z8 = {0, 0, 0, 0, 0, 0, 0, 0};
  __builtin_amdgcn_tensor_load_to_lds(g0, g1, z4, z4, z8, 0);  // clang-23 / therock form
#else
  __builtin_amdgcn_tensor_load_to_lds(g0, g1, z4, z4, 0);      // ROCm 7.2 clang-22 form
#endif
}

// ---------------------------------------------------------------------------
// Double-buffered TDM + WMMA GEMM.
//   C[M,NCOLS] (+bias, act) = A[M,K](lda) @ B[K,NCOLS](row-major, ldb=NCOLS)
// Block = 256 threads (8 waves) computes BROWS x NCOLS of C.
//   RT = BROWS/16 row tiles; CG = 8/RT column groups; CT col tiles per wave.
// K must be a multiple of CHUNK(=32).  act: 0 none,1 leaky,2 PReLU,3 relu.
// ---------------------------------------------------------------------------
template <int BROWS, int NCOLS, int CT>
__global__ void gemm_tdm_wmma(
    const float* __restrict__ A, int lda,
    const float* __restrict__ B,
    const float* __restrict__ bias,
    const float* __restrict__ prelu,
    float* __restrict__ C, int ldc,
    int M, int K, int act)
{
  constexpr int CHUNK = 32;
  constexpr int RT = BROWS / 16;
  constexpr int CG = 8 / RT;
  constexpr int WCOLS = NCOLS / CG;        // == CT*16
  __shared__ float Ash[2][BROWS * CHUNK];
  __shared__ float Bsh[2][CHUNK * NCOLS];

  const int lane = threadIdx.x & 31;
  const int wave = threadIdx.x >> 5;
  const int half = lane >> 4;
  const int l16  = lane & 15;
  const int row_tile = wave / CG;
  const int col_grp  = (wave % CG) * WCOLS;
  const int rows0 = blockIdx.x * BROWS;
  const int lrow  = row_tile * 16 + l16;

  const uint32_t ashAddr[2] = {(uint32_t)(uintptr_t)&Ash[0][0],
                               (uint32_t)(uintptr_t)&Ash[1][0]};
  const uint32_t bshAddr[2] = {(uint32_t)(uintptr_t)&Bsh[0][0],
                               (uint32_t)(uintptr_t)&Bsh[1][0]};

  v8f acc[CT] = {};
  const int nch = K / CHUNK;

  auto issue = [&](int c, int buf) {
    const int kc = c * CHUNK;
    // A tile: BROWS x CHUNK at (rows0, kc); OOB rows/cols hardware zero-filled
    tdm_load_2d_f32(ashAddr[buf], A + (int64_t)rows0 * lda + kc,
                    CHUNK, BROWS, (uint32_t)(K - kc), (uint32_t)(M - rows0),
                    (uint64_t)lda);
    // B panel: CHUNK x NCOLS at row kc
    tdm_load_2d_f32(bshAddr[buf], B + (int64_t)kc * NCOLS,
                    NCOLS, CHUNK, (uint32_t)NCOLS, (uint32_t)(K - kc),
                    (uint64_t)NCOLS);
  };

  if (wave == 0) issue(0, 0);              // preload chunk 0

  for (int c = 0; c < nch; ++c) {
    const int buf = c & 1;
    if (wave == 0) {                       // wave-uniform
      if (c + 1 < nch) {
        issue(c + 1, buf ^ 1);             // overlap next DMA with this compute
        __builtin_amdgcn_s_wait_tensorcnt(2);  // in-order: current buffer landed
      } else {
        __builtin_amdgcn_s_wait_tensorcnt(0);
      }
    }
    __syncthreads();

    const float* As = &Ash[buf][0];
    const float* Bs = &Bsh[buf][0];
#pragma unroll
    for (int kk = 0; kk < CHUNK; kk += 4) {
      const int ka = kk + 2 * half;
      v2f a;
      a.x = As[lrow * CHUNK + ka];
      a.y = As[lrow * CHUNK + ka + 1];
      const float* br0 = &Bs[ka * NCOLS + col_grp + l16];
      const float* br1 = &Bs[(ka + 1) * NCOLS + col_grp + l16];
#pragma unroll
      for (int ct = 0; ct < CT; ++ct) {
        v2f b;
        b.x = br0[ct * 16];
        b.y = br1[ct * 16];
        acc[ct] = __builtin_amdgcn_wmma_f32_16x16x4_f32(false, a, false, b,
                                                        (short)0, acc[ct], false, false);
      }
    }
    __syncthreads();                       // all reads done before buffer reuse
  }

  // epilogue: C VGPR j holds rows j (lanes 0-15) / j+8 (lanes 16-31)
#pragma unroll
  for (int ct = 0; ct < CT; ++ct) {
    const int colB = col_grp + ct * 16 + l16;
    const float bv = bias ? bias[colB] : 0.0f;
    const float pa = (act == 2) ? prelu[colB] : 0.0f;
#pragma unroll
    for (int j = 0; j < 8; ++j) {
      const int r = rows0 + row_tile * 16 + j + 8 * half;
      if (r < M) {
        float v = acc[ct][j] + bv;
        if (act == 1)      v = (v > 0.0f) ? v : LEAKY_SLOPE * v;
        else if (act == 2) v = (v > 0.0f) ? v : pa * v;
        else if (act == 3) v = fmaxf(v, 0.0f);
        C[(int64_t)r * ldc + colB] = v;
      }
    }
  }
}

// ---------------------------------------------------------------------------
// Generic strided WMMA GEMM (only for tiny-K encoder slices: K=6, K=2, N=32).
// ---------------------------------------------------------------------------
__global__ void gemm16x16_wmma_f32(
    const float* __restrict__ A, int lda,
    const float* __restrict__ B, int ldb,
    const float* __restrict__ bias,
    const float* __restrict__ prelu,
    float* __restrict__ C, int ldc,
    int M, int N, int K, int act)
{
  const int wave = (int)(((int64_t)blockIdx.x * blockDim.x + threadIdx.x) >> 5);
  const int lane = threadIdx.x & 31;
  const int tilesN = N >> 4;
  const int tm = wave / tilesN;
  const int tn = wave - tm * tilesN;
  if (tm * 16 >= M) return;                // wave-uniform

  const int half = lane >> 4;
  const int l16  = lane & 15;
  int rowA = tm * 16 + l16;
  if (rowA >= M) rowA = M - 1;
  const int colB = tn * 16 + l16;

  const float* Arow = A + (int64_t)rowA * lda;
  v8f acc = {};

  int k = 0;
  const int Kfast = K & ~3;
  for (; k < Kfast; k += 4) {
    const int ka = k + 2 * half;
    v2f a, b;
    a.x = Arow[ka];
    a.y = Arow[ka + 1];
    b.x = B[(int64_t)ka * ldb + colB];
    b.y = B[(int64_t)(ka + 1) * ldb + colB];
    acc = __builtin_amdgcn_wmma_f32_16x16x4_f32(false, a, false, b, (short)0, acc, false, false);
  }
  if (k < K) {
    const int ka = k + 2 * half;
    v2f a = {0.0f, 0.0f}, b = {0.0f, 0.0f};
    if (ka < K)     { a.x = Arow[ka];     b.x = B[(int64_t)ka * ldb + colB]; }
    if (ka + 1 < K) { a.y = Arow[ka + 1]; b.y = B[(int64_t)(ka + 1) * ldb + colB]; }
    acc = __builtin_amdgcn_wmma_f32_16x16x4_f32(false, a, false, b, (short)0, acc, false, false);
  }

  const float bv = bias ? bias[colB] : 0.0f;
  const float pa = (act == 2) ? prelu[colB] : 0.0f;
#pragma unroll
  for (int j = 0; j < 8; ++j) {
    const int r = tm * 16 + j + 8 * half;
    if (r < M) {
      float v = acc[j] + bv;
      if (act == 1)      v = (v > 0.0f) ? v : LEAKY_SLOPE * v;
      else if (act == 2) v = (v > 0.0f) ? v : pa * v;
      else if (act == 3) v = fmaxf(v, 0.0f);
      C[(int64_t)r * ldc + colB] = v;
    }
  }
}

// ---------------------------------------------------------------------------
// Sparse / elementwise helpers
// ---------------------------------------------------------------------------
__global__ void gather_rows128(const float* __restrict__ src, const int* __restrict__ idx,
                               float* __restrict__ dst, int n)
{
  int64_t t = (int64_t)blockIdx.x * blockDim.x + threadIdx.x;
  int i = (int)(t >> 5);
  if (i >= n) return;
  int lane = (int)(t & 31);
  const float4* s = (const float4*)(src + (int64_t)idx[i] * 128);
  float4* d = (float4*)(dst + (int64_t)i * 128);
  d[lane] = s[lane];
}

__global__ void fill_f32(float* __restrict__ p, float v, int n)
{
  int i = blockIdx.x * blockDim.x + threadIdx.x;
  if (i < n) p[i] = v;
}

__global__ void deg_accum(const int* __restrict__ dst, float* __restrict__ deg, int E)
{
  int e = blockIdx.x * blockDim.x + threadIdx.x;
  if (e < E) atomicAdd(&deg[dst[e]], 1.0f);
}

__global__ void rsqrt_inplace(float* __restrict__ p, int n)
{
  int i = blockIdx.x * blockDim.x + threadIdx.x;
  if (i < n) p[i] = rsqrtf(p[i]);          // deg >= 1 (self loop)
}

__global__ void spmm_edges128(const int* __restrict__ src, const int* __restrict__ dst,
                              const float* __restrict__ dinv, const float* __restrict__ h,
                              float* __restrict__ out, int E)
{
  int64_t t = (int64_t)blockIdx.x * blockDim.x + threadIdx.x;
  int e = (int)(t >> 5);
  if (e >= E) return;
  int lane = (int)(t & 31);
  int s = src[e], d = dst[e];
  float w = dinv[s] * dinv[d];
  float4 v = ((const float4*)(h + (int64_t)s * 128))[lane];
  float* o = out + (int64_t)d * 128 + lane * 4;
  atomicAdd(o + 0, v.x * w);
  atomicAdd(o + 1, v.y * w);
  atomicAdd(o + 2, v.z * w);
  atomicAdd(o + 3, v.w * w);
}

__global__ void gcn_finalize128(float* __restrict__ agg, const float* __restrict__ h,
                                const float* __restrict__ dinv, const float* __restrict__ bias,
                                int n)
{
  int64_t t = (int64_t)blockIdx.x * blockDim.x + threadIdx.x;
  int i = (int)(t >> 5);
  if (i >= n) return;
  int lane = (int)(t & 31);
  float di = dinv[i];
  float w = di * di;
  float4 hv = ((const float4*)(h + (int64_t)i * 128))[lane];
  float4 bv = ((const float4*)bias)[lane];
  float4* ap = (float4*)(agg + (int64_t)i * 128);
  float4 av = ap[lane];
  av.x = fmaxf(av.x + hv.x * w + bv.x, 0.0f);
  av.y = fmaxf(av.y + hv.y * w + bv.y, 0.0f);
  av.z = fmaxf(av.z + hv.z * w + bv.z, 0.0f);
  av.w = fmaxf(av.w + hv.w * w + bv.w, 0.0f);
  ap[lane] = av;
}

__global__ void scatter_rows128(const float* __restrict__ x, const int* __restrict__ seg,
                                float* __restrict__ out, int n, int outStride, int colOff)
{
  int64_t t = (int64_t)blockIdx.x * blockDim.x + threadIdx.x;
  int i = (int)(t >> 5);
  if (i >= n) return;
  int lane = (int)(t & 31);
  int p = seg[i];
  float4 v = ((const float4*)(x + (int64_t)i * 128))[lane];
  float* o = out + (int64_t)p * outStride + colOff + lane * 4;
  atomicAdd(o + 0, v.x);
  atomicAdd(o + 1, v.y);
  atomicAdd(o + 2, v.z);
  atomicAdd(o + 3, v.w);
}

__global__ void relu_inplace(float* __restrict__ p, int64_t n)
{
  int64_t i = (int64_t)blockIdx.x * blockDim.x + threadIdx.x;
  if (i < n) p[i] = fmaxf(p[i], 0.0f);
}

// ---------------------------------------------------------------------------

static inline void launch_gemm_generic(const float* A, int lda, const float* B, int ldb,
                                       const float* bias, const float* prelu,
                                       float* C, int ldc, int M, int N, int K, int act,
                                       hipStream_t s)
{
  int tilesM = (M + 15) / 16;
  int tilesN = N / 16;
  int64_t waves = (int64_t)tilesM * tilesN;
  int blocks = (int)((waves + 7) / 8);
  gemm16x16_wmma_f32<<<blocks, 256, 0, s>>>(A, lda, B, ldb, bias, prelu,
                                            C, ldc, M, N, K, act);
}

// N=128 path: 64-row blocks, 4 col tiles / wave
static inline void launch_gemm_n128(const float* A, int lda, const float* B,
                                    const float* bias, const float* prelu,
                                    float* C, int M, int K, int act, hipStream_t s)
{
  int blocks = (M + 63) / 64;
  gemm_tdm_wmma<64, 128, 4><<<blocks, 256, 0, s>>>(A, lda, B, bias, prelu,
                                                   C, 128, M, K, act);
}

// N=32 path (encoder K=768 slices): 128-row blocks, 2 col tiles / wave
static inline void launch_gemm_n32(const float* A, int lda, const float* B,
                                   const float* bias, float* C, int ldc,
                                   int M, int K, int act, hipStream_t s)
{
  int blocks = (M + 127) / 128;
  gemm_tdm_wmma<128, 32, 2><<<blocks, 256, 0, s>>>(A, lda, B, bias, nullptr,
                                                   C, ldc, M, K, act);
}

static inline int gridRows(int64_t rows) { return (int)((rows * 32 + 255) / 256); }

extern "C" void kernel_launch(void* const* d_in, const int* in_sizes, int n_in,
                              void* d_out, int out_size, void* d_ws, size_t ws_size,
                              hipStream_t stream)
{
  (void)in_sizes; (void)n_in; (void)out_size; (void)ws_size;

  const float* x_raw   = (const float*)d_in[0];
  const float* W_des   = (const float*)d_in[1];
  const float* b_des   = (const float*)d_in[2];
  const float* W_tw    = (const float*)d_in[3];
  const float* b_tw    = (const float*)d_in[4];
  const float* W_num   = (const float*)d_in[5];
  const float* b_num   = (const float*)d_in[6];
  const float* W_cat   = (const float*)d_in[7];
  const float* b_cat   = (const float*)d_in[8];
  const float* W_in    = (const float*)d_in[9];
  const float* b_in    = (const float*)d_in[10];
  const float* prelu_a = (const float*)d_in[11];
  const float* W_conv  = (const float*)d_in[12];
  const float* b_conv  = (const float*)d_in[13];
  const float* Wc1     = (const float*)d_in[14];
  const float* bc1     = (const float*)d_in[15];
  const float* Wc2     = (const float*)d_in[16];
  const float* bc2     = (const float*)d_in[17];
  const int* nodelist  = (const int*)d_in[18];
  const int* e0        = (const int*)d_in[19];
  const int* e1        = (const int*)d_in[20];
  const int* e2        = (const int*)d_in[21];
  const int* parent1   = (const int*)d_in[22];
  const int* parent2   = (const int*)d_in[23];
  const int* batch1    = (const int*)d_in[24];
  const int* batch2    = (const int*)d_in[25];
  float* out = (float*)d_out;

  constexpr int NU = 50000, N0 = 200000, N1 = 50000, N2 = 12500;
  constexpr int E0 = 1600000, E1 = 400000, E2 = 100000;
  constexpr int B = 128;

  // ---- workspace layout (floats) ----
  float* ws     = (float*)d_ws;
  float* feat   = ws;                          // NU*128
  float* enc    = feat + (int64_t)NU * 128;    // NU*128
  float* x0     = enc  + (int64_t)NU * 128;    // N0*128 (reused as agg0 output)
  float* h0     = x0   + (int64_t)N0 * 128;    // N0*128
  float* x1     = h0   + (int64_t)N0 * 128;    // N1*128  -> xs[0], kept
  float* h1     = x1   + (int64_t)N1 * 128;    // N1*128
  float* y1     = h1   + (int64_t)N1 * 128;    // N1*128
  float* x2     = y1   + (int64_t)N1 * 128;    // N2*128  -> xs[1], kept
  float* h2     = x2   + (int64_t)N2 * 128;    // N2*128
  float* y2     = h2   + (int64_t)N2 * 128;    // N2*128  -> xs[2], kept
  float* deg    = y2   + (int64_t)N2 * 128;    // max N0
  float* pooled = deg  + N0;                   // B*384
  float* z      = pooled + (int64_t)B * 384;   // B*128

  // ---- feature encoder into feat = [d|t|n|c] ----
  // x_raw row layout: num[0:6] | tweet[6:774] | cat[774:776] | des[776:1544]
  launch_gemm_n32(x_raw + 776, 1544, W_des, b_des, feat + 0,  128, NU, 768, 1, stream);
  launch_gemm_n32(x_raw + 6,   1544, W_tw,  b_tw,  feat + 32, 128, NU, 768, 1, stream);
  launch_gemm_generic(x_raw + 0,   1544, W_num, 32, b_num, nullptr, feat + 64, 128, NU, 32, 6, 1, stream);
  launch_gemm_generic(x_raw + 774, 1544, W_cat, 32, b_cat, nullptr, feat + 96, 128, NU, 32, 2, 1, stream);
  // enc = PReLU(feat @ W_in + b_in)
  launch_gemm_n128(feat, 128, W_in, b_in, prelu_a, enc, NU, 128, 2, stream);

  // ---- x0 = enc[nodelist] ----
  gather_rows128<<<gridRows(N0), 256, 0, stream>>>(enc, nodelist, x0, N0);

  // ---- GCN layer: lin -> sym-norm agg (self loops) -> +bias -> relu ----
  auto gcn_layer = [&](const float* xin, float* hlin, float* yout,
                       const int* ei, int E, int n, int layer) {
    fill_f32<<<(n + 255) / 256, 256, 0, stream>>>(deg, 1.0f, n);
    deg_accum<<<(E + 255) / 256, 256, 0, stream>>>(ei + E, deg, E);
    rsqrt_inplace<<<(n + 255) / 256, 256, 0, stream>>>(deg, n);
    launch_gemm_n128(xin, 128, W_conv + (int64_t)layer * 128 * 128,
                     nullptr, nullptr, hlin, n, 128, 0, stream);
    hipMemsetAsync(yout, 0, (size_t)n * 128 * sizeof(float), stream);
    spmm_edges128<<<gridRows(E), 256, 0, stream>>>(ei, ei + E, deg, hlin, yout, E);
    gcn_finalize128<<<gridRows(n), 256, 0, stream>>>(yout, hlin, deg,
                                                     b_conv + layer * 128, n);
  };

  gcn_layer(x0, h0, x0, e0, E0, N0, 0);
  hipMemsetAsync(x1, 0, (size_t)N1 * 128 * sizeof(float), stream);
  scatter_rows128<<<gridRows(N0), 256, 0, stream>>>(x0, parent1, x1, N0, 128, 0);
  relu_inplace<<<(int)(((int64_t)N1 * 128 + 255) / 256), 256, 0, stream>>>(x1, (int64_t)N1 * 128);

  gcn_layer(x1, h1, y1, e1, E1, N1, 1);
  hipMemsetAsync(x2, 0, (size_t)N2 * 128 * sizeof(float), stream);
  scatter_rows128<<<gridRows(N1), 256, 0, stream>>>(y1, parent2, x2, N1, 128, 0);
  relu_inplace<<<(int)(((int64_t)N2 * 128 + 255) / 256), 256, 0, stream>>>(x2, (int64_t)N2 * 128);

  gcn_layer(x2, h2, y2, e2, E2, N2, 2);

  // ---- global sum pooling into [B, 384] ----
  hipMemsetAsync(pooled, 0, (size_t)B * 384 * sizeof(float), stream);
  scatter_rows128<<<gridRows(N1), 256, 0, stream>>>(x1, batch1, pooled, N1, 384, 0);
  scatter_rows128<<<gridRows(N2), 256, 0, stream>>>(x2, batch2, pooled, N2, 384, 128);
  scatter_rows128<<<gridRows(N2), 256, 0, stream>>>(y2, batch2, pooled, N2, 384, 256);

  // ---- head: z = relu(pooled @ Wc1 + bc1); out = z @ Wc2 + bc2 ----
  launch_gemm_n128(pooled, 384, Wc1, bc1, nullptr, z, B, 384, 3, stream);
  launch_gemm_n128(z, 128, Wc2, bc2, nullptr, out, B, 128, 0, stream);
}